// DynamicConv_27178553049156
// MI455X (gfx1250) — compile-verified
//
#include <hip/hip_runtime.h>

typedef float v2f __attribute__((ext_vector_type(2)));
typedef float v8f __attribute__((ext_vector_type(8)));

#define L_LEN 2048
#define D_DIM 128
#define E_DIM 128
#define F_DIM 8
#define K1_T 3
#define K2_T 3
#define T_W 72                    // 3*8*3 weighted terms
#define T_TOTAL 75                // + 3 bias terms (f-weight == 1)
#define K_TOT (T_TOTAL * D_DIM)   // 9600
#define FPAD_ROWS (L_LEN + 2)     // 2050 (zero rows at both ends)
#define FPAD_FLOATS 16512         // padded allocation
#define W2_FLOATS (K_TOT * E_DIM) // 1228800
#define LTILE 64                  // l rows per workgroup
#define OUT_N (L_LEN * E_DIM)     // 262144
#define TSPLIT 5
#define TPER 15                   // 75 / 5
#define NTHREADS 128              // 4 waves: 2 l-groups x 2 e-groups

// ---------------------------------------------------------------------------
// Kernel 1: feature projection f = u @ proj with one zero halo row on each
// end: fpad[row] corresponds to sequence position l = row - 1.
// ---------------------------------------------------------------------------
__global__ void featproj_kernel(const float* __restrict__ u,
                                const float* __restrict__ proj,
                                float* __restrict__ fpad) {
    int idx = blockIdx.x * blockDim.x + threadIdx.x;
    if (idx >= FPAD_ROWS * F_DIM) return;
    int f   = idx & (F_DIM - 1);
    int row = idx >> 3;
    int l   = row - 1;
    float acc = 0.0f;
    if (l >= 0 && l < L_LEN) {
        const float* up = u + (long)l * D_DIM;
#pragma unroll 8
        for (int d = 0; d < D_DIM; ++d)
            acc = fmaf(up[d], proj[d * F_DIM + f], acc);
    }
    fpad[row * F_DIM + f] = acc;
}

// ---------------------------------------------------------------------------
// Kernel 2: pack conv weights + bias into W_big[k=9600][e=128].
// K ordering: k = t*128 + d, t = (k1*8 + f)*3 + k2 for t<72, t-72 = k2 (bias).
// Storage interleaves k-pairs so a lane's (K=2p, K=2p+1) B fragment at column
// e is one contiguous aligned float2: W2[((k>>1)*128 + e)*2 + (k&1)].
// ---------------------------------------------------------------------------
__global__ void weight_pack_kernel(const float* __restrict__ conv_w,
                                   const float* __restrict__ conv_b,
                                   float* __restrict__ W2) {
    long idx = (long)blockIdx.x * blockDim.x + threadIdx.x;
    if (idx >= (long)K_TOT * E_DIM) return;
    int e = (int)(idx & (E_DIM - 1));
    int k = (int)(idx >> 7);
    int d = k & (D_DIM - 1);
    int t = k >> 7;
    float v;
    if (t < T_W) {
        int k1 = t / 24;
        int r  = t - k1 * 24;
        int fi = r / 3;
        int k2 = r - fi * 3;
        // conv_w layout [E][K1][F][M], m = d*3 + k2
        v = conv_w[(((long)e * K1_T + k1) * F_DIM + fi) * (K2_T * D_DIM) + d * K2_T + k2];
    } else {
        int k2 = t - T_W;
        v = conv_b[(long)e * (K2_T * D_DIM) + d * K2_T + k2];
    }
    W2[(((long)(k >> 1)) * E_DIM + e) * 2 + (k & 1)] = v;
}

// ---------------------------------------------------------------------------
// Kernel 3: fused dynamic-conv as implicit GEMM, 2x4 register tiling.
// Workgroup = 64 l-rows x 128 e, 4 waves. Wave w: lp = w&1 -> row base 32*lp
// (2 l-subtiles), ep = w>>1 -> e base 64*ep (4 e-tiles). grid.y splits K(t).
// Per inner step: 2 A fragments (2 ds_b64 + 4 mul) + 4 B loads (4 global_b64)
// feed 8 v_wmma_f32_16x16x4_f32 -> 0.25 ds + 0.5 mul + 0.5 vmem per WMMA.
// ---------------------------------------------------------------------------
__global__ void __launch_bounds__(NTHREADS)
dynconv_wmma_kernel(const float* __restrict__ u,
                    const float* __restrict__ fpad,
                    const float* __restrict__ W2,
                    float* __restrict__ dst,
                    int t_per) {
    __shared__ float Ush[LTILE + 2][132];   // 66 halo'd u rows, padded stride
    __shared__ float Fsh[LTILE + 2][F_DIM]; // 66 halo'd f rows

    const int tid = threadIdx.x;
    const int l0  = blockIdx.x * LTILE;
    const int tz  = blockIdx.y;
    const int t0  = tz * t_per;
    const int t1  = (t0 + t_per < T_TOTAL) ? (t0 + t_per) : T_TOTAL;
    dst += (long)tz * OUT_N;

    // stage u rows l0-1 .. l0+64 (zero pad out of range)
    for (int i = tid; i < (LTILE + 2) * D_DIM; i += NTHREADS) {
        int row = i >> 7, d = i & 127;
        int l = l0 - 1 + row;
        Ush[row][d] = (l >= 0 && l < L_LEN) ? u[(long)l * D_DIM + d] : 0.0f;
    }
    // stage f rows (fpad already halo-padded: fpad row = l + 1)
    for (int i = tid; i < (LTILE + 2) * F_DIM; i += NTHREADS) {
        int row = i >> 3, f = i & 7;
        Fsh[row][f] = fpad[(l0 + row) * F_DIM + f];
    }
    __syncthreads();

    const int lane = tid & 31;
    const int wave = tid >> 5;
    const int m    = lane & 15;        // M index within tile (also N for B/D)
    const int h    = lane >> 4;        // K half-select
    const int r0   = (wave & 1) * 32;  // 2 l-subtiles: r0, r0+16
    const int e0   = (wave >> 1) * 64; // 4 e-tiles: e0, e0+16, e0+32, e0+48

    v8f a00 = {0.f,0.f,0.f,0.f,0.f,0.f,0.f,0.f};
    v8f a01 = a00, a02 = a00, a03 = a00;
    v8f a10 = a00, a11 = a00, a12 = a00, a13 = a00;

    for (int t = t0; t < t1; ++t) {
        float s0, s1;
        int k2;
        if (t < T_W) {
            int k1 = t / 24;
            int r  = t - k1 * 24;
            int fi = r / 3;
            k2 = r - fi * 3;
            s0 = Fsh[r0 + m + k1][fi];          // f[l0+r0+m + k1 - 1, fi]
            s1 = Fsh[r0 + 16 + m + k1][fi];
        } else {
            k2 = t - T_W;
            s0 = s1 = 1.0f;                      // bias terms
        }
        const float* Ur0 = &Ush[r0 + m + k2][2 * h];
        const float* Ur1 = &Ush[r0 + 16 + m + k2][2 * h];
        const float* Bb  = W2 + (((long)t * 64 + h) * E_DIM + e0 + m) * 2;

#pragma unroll 2
        for (int dc = 0; dc < D_DIM / 4; ++dc) {
            v2f fa0, fa1;
            fa0.x = s0 * Ur0[4 * dc];            // A[m, kbase + 2h]
            fa0.y = s0 * Ur0[4 * dc + 1];        // A[m, kbase + 2h + 1]
            fa1.x = s1 * Ur1[4 * dc];
            fa1.y = s1 * Ur1[4 * dc + 1];
            const float* Bc = Bb + (long)dc * 2 * E_DIM * 2;
            v2f b0 = *(const v2f*)(Bc);
            v2f b1 = *(const v2f*)(Bc + 32);     // +16 e columns
            v2f b2 = *(const v2f*)(Bc + 64);
            v2f b3 = *(const v2f*)(Bc + 96);
            a00 = __builtin_amdgcn_wmma_f32_16x16x4_f32(false, fa0, false, b0, (short)0, a00, false, false);
            a01 = __builtin_amdgcn_wmma_f32_16x16x4_f32(false, fa0, false, b1, (short)0, a01, false, false);
            a02 = __builtin_amdgcn_wmma_f32_16x16x4_f32(false, fa0, false, b2, (short)0, a02, false, false);
            a03 = __builtin_amdgcn_wmma_f32_16x16x4_f32(false, fa0, false, b3, (short)0, a03, false, false);
            a10 = __builtin_amdgcn_wmma_f32_16x16x4_f32(false, fa1, false, b0, (short)0, a10, false, false);
            a11 = __builtin_amdgcn_wmma_f32_16x16x4_f32(false, fa1, false, b1, (short)0, a11, false, false);
            a12 = __builtin_amdgcn_wmma_f32_16x16x4_f32(false, fa1, false, b2, (short)0, a12, false, false);
            a13 = __builtin_amdgcn_wmma_f32_16x16x4_f32(false, fa1, false, b3, (short)0, a13, false, false);
        }
        // prefetch next t's B block (streams W2 through L2)
        if (t + 1 < t1)
            __builtin_prefetch(Bb + (long)64 * E_DIM * 2, 0, 1);
    }

    // D layout: VGPR r <-> M = r + 8*h, N = m
#pragma unroll
    for (int li = 0; li < 2; ++li) {
        v8f v0 = li ? a10 : a00;
        v8f v1 = li ? a11 : a01;
        v8f v2 = li ? a12 : a02;
        v8f v3 = li ? a13 : a03;
        float* o = dst + (long)(l0 + r0 + 16 * li) * E_DIM + e0 + m;
#pragma unroll
        for (int r = 0; r < 8; ++r) {
            long off = (long)(r + 8 * h) * E_DIM;
            o[off]      = v0[r];
            o[off + 16] = v1[r];
            o[off + 32] = v2[r];
            o[off + 48] = v3[r];
        }
    }
}

// ---------------------------------------------------------------------------
// Kernel 4: reduce the TSPLIT K-partials into the final output.
// ---------------------------------------------------------------------------
__global__ void reduce_kernel(const float* __restrict__ partial,
                              float* __restrict__ out) {
    int i = blockIdx.x * blockDim.x + threadIdx.x;
    if (i >= OUT_N) return;
    float s = 0.0f;
#pragma unroll
    for (int z = 0; z < TSPLIT; ++z)
        s += partial[(long)z * OUT_N + i];
    out[i] = s;
}

// ---------------------------------------------------------------------------
extern "C" void kernel_launch(void* const* d_in, const int* in_sizes, int n_in,
                              void* d_out, int out_size, void* d_ws, size_t ws_size,
                              hipStream_t stream) {
    const float* u      = (const float*)d_in[0];
    const float* proj   = (const float*)d_in[1];
    const float* conv_w = (const float*)d_in[2];
    const float* conv_b = (const float*)d_in[3];
    float* out = (float*)d_out;

    float* ws      = (float*)d_ws;
    float* fpad    = ws;                        // FPAD_FLOATS floats
    float* W2      = ws + FPAD_FLOATS;          // W2_FLOATS floats (~4.7 MB)
    float* partial = W2 + W2_FLOATS;            // TSPLIT * OUT_N floats (~5 MB)

    featproj_kernel<<<(FPAD_ROWS * F_DIM + 255) / 256, 256, 0, stream>>>(u, proj, fpad);
    weight_pack_kernel<<<((long)K_TOT * E_DIM + 255) / 256, 256, 0, stream>>>(conv_w, conv_b, W2);

    const size_t need_split =
        (size_t)(FPAD_FLOATS + W2_FLOATS + (size_t)TSPLIT * OUT_N) * sizeof(float);
    if (ws_size >= need_split) {
        // K-split across grid.y for wave-level parallelism; deterministic
        // (private partials + fixed-order serial sum, no float atomics).
        dynconv_wmma_kernel<<<dim3(L_LEN / LTILE, TSPLIT), NTHREADS, 0, stream>>>(
            u, fpad, W2, partial, TPER);
        reduce_kernel<<<(OUT_N + 255) / 256, 256, 0, stream>>>(partial, out);
    } else {
        dynconv_wmma_kernel<<<dim3(L_LEN / LTILE, 1), NTHREADS, 0, stream>>>(
            u, fpad, W2, out, T_TOTAL);
    }
}